// GAT_43112881717973
// MI455X (gfx1250) — compile-verified
//
#include <hip/hip_runtime.h>
#include <float.h>

typedef float v2f __attribute__((ext_vector_type(2)));
typedef float v8f __attribute__((ext_vector_type(8)));

// ---- order-preserving float<->uint key for atomic segment-max ------------
__device__ __forceinline__ unsigned f32_key(float f) {
  unsigned u = __float_as_uint(f);
  return (u & 0x80000000u) ? ~u : (u | 0x80000000u);
}
__device__ __forceinline__ float f32_unkey(unsigned k) {
  return (k & 0x80000000u) ? __uint_as_float(k ^ 0x80000000u)
                           : __uint_as_float(~k);
}

// ---- per-layer init: acc = 0, denom = 0, maxkey = 0 (== -inf) ------------
__global__ void gat_init(float* __restrict__ acc, float* __restrict__ denom,
                         unsigned* __restrict__ mbits, int N, int C) {
  const int i = blockIdx.x * blockDim.x + threadIdx.x;
  if (i < N * C) acc[i] = 0.0f;
  if (i < N) { denom[i] = 0.0f; mbits[i] = 0u; }
}

// ---- H = X @ W via V_WMMA_F32_16X16X4_F32 --------------------------------
// One wave computes a full 16 x C row strip: A fragment loaded once per
// k-step feeds CT column-tile WMMAs (A reuse => minimal X traffic).
// Branchless column masking keeps EXEC all-ones across every WMMA.
// A frag (16x4 f32, 2 VGPR): lane L holds row M=L&15; v0=K[half*2], v1=K[half*2+1]
// B frag (4x16 f32, 2 VGPR): lane L holds col N=L&15; v0=row K[half*2], v1=K[half*2+1]
// C/D   (16x16 f32, 8 VGPR): vr, lanes 0-15 -> M=r, lanes 16-31 -> M=r+8
template <int K, int C>
__global__ void gat_gemm_wmma(const float* __restrict__ X, const float* __restrict__ W,
                              float* __restrict__ H, int N) {
  constexpr int CT = (C + 15) / 16;                 // column tiles (4 or 3)
  const int lane = threadIdx.x & 31;
  const int rt   = (blockIdx.x * blockDim.x + threadIdx.x) >> 5;
  const int rtiles = (N + 15) >> 4;
  if (rt >= rtiles) return;                         // wave-uniform exit
  const int row0 = rt << 4;
  const int m    = lane & 15;
  const int half = lane >> 4;                       // 0: K+0/1 , 1: K+2/3
  const int arow = min(row0 + m, N - 1);            // clamp keeps lanes active
  const float* __restrict__ xrow = X + (size_t)arow * K;

  v8f acc[CT];
#pragma unroll
  for (int ct = 0; ct < CT; ++ct) acc[ct] = (v8f){0.f,0.f,0.f,0.f,0.f,0.f,0.f,0.f};

#pragma unroll
  for (int k0 = 0; k0 < K; k0 += 4) {
    const int ka = k0 + (half << 1);
    const v2f a = *(const v2f*)(xrow + ka);         // 8B-aligned (ka even)
#pragma unroll
    for (int ct = 0; ct < CT; ++ct) {
      const int col = (ct << 4) + m;
      float msk; int cc;
      if ((ct + 1) * 16 <= C) { msk = 1.0f; cc = col; }           // full tile (folded)
      else { msk = (col < C) ? 1.0f : 0.0f; cc = (col < C) ? col : (C - 1); }
      v2f b;
      b.x = W[(size_t)ka * C + cc] * msk;
      b.y = W[(size_t)(ka + 1) * C + cc] * msk;
      acc[ct] = __builtin_amdgcn_wmma_f32_16x16x4_f32(false, a, false, b,
                                                      (short)0, acc[ct], false, false);
    }
  }

#pragma unroll
  for (int ct = 0; ct < CT; ++ct) {
    const int col = (ct << 4) + m;
#pragma unroll
    for (int r = 0; r < 8; ++r) {
      const int row = row0 + r + (half << 3);
      if (row < N && col < C) H[(size_t)row * C + col] = acc[ct][r];
    }
  }
}

// ---- alpha_s[n] = h[n].a_src ; alpha_d[n] = h[n].a_dst (one wave / node) -
__global__ void gat_alphas(const float* __restrict__ H, const float* __restrict__ a_src,
                           const float* __restrict__ a_dst, float* __restrict__ as_,
                           float* __restrict__ ad_, int N, int C) {
  const int lane = threadIdx.x & 31;
  const int n = (blockIdx.x * blockDim.x + threadIdx.x) >> 5;
  if (n >= N) return;
  float ss = 0.f, sd = 0.f;
  for (int f = lane; f < C; f += 32) {
    const float h = H[(size_t)n * C + f];
    ss += h * a_src[f];
    sd += h * a_dst[f];
  }
  for (int o = 16; o > 0; o >>= 1) {
    ss += __shfl_xor(ss, o, 32);
    sd += __shfl_xor(sd, o, 32);
  }
  if (lane == 0) { as_[n] = ss; ad_[n] = sd; }
}

// ---- pass 1: e = leaky_relu(as[src]+ad[dst]); segment-max into mbits -----
__global__ void gat_edge_max(const long long* __restrict__ ei, int E, int N,
                             const float* __restrict__ as_, const float* __restrict__ ad_,
                             float* __restrict__ ebuf, unsigned* __restrict__ mbits) {
  const int e = blockIdx.x * blockDim.x + threadIdx.x;
  if (e >= E + N) return;
  int s, d;
  if (e < E) { s = (int)ei[e]; d = (int)ei[(size_t)E + e]; }
  else       { s = d = e - E; }                       // virtual self-loop
  float v = as_[s] + ad_[d];
  v = (v > 0.f) ? v : 0.2f * v;                       // leaky_relu, slope 0.2
  ebuf[e] = v;
  atomicMax(mbits + d, f32_key(v));
}

// ---- pass 2: ex = exp(e - max[dst]); denom[dst] += ex --------------------
__global__ void gat_edge_exp(const long long* __restrict__ ei, int E, int N,
                             const unsigned* __restrict__ mbits,
                             float* __restrict__ ebuf, float* __restrict__ denom) {
  const int e = blockIdx.x * blockDim.x + threadIdx.x;
  if (e >= E + N) return;
  const int d = (e < E) ? (int)ei[(size_t)E + e] : e - E;
  const float ex = __expf(ebuf[e] - f32_unkey(mbits[d]));
  ebuf[e] = ex;
  atomicAdd(denom + d, ex);
}

// ---- pass 3: acc[dst] += (ex/denom[dst]) * h[src]  (one wave / edge) -----
__global__ void gat_edge_agg(const long long* __restrict__ ei, int E, int N,
                             const float* __restrict__ ebuf, const float* __restrict__ denom,
                             const float* __restrict__ H, float* __restrict__ acc, int C) {
  const int lane = threadIdx.x & 31;
  const int e = (blockIdx.x * blockDim.x + threadIdx.x) >> 5;
  if (e >= E + N) return;
  int s, d;
  if (e < E) { s = (int)ei[e]; d = (int)ei[(size_t)E + e]; }
  else       { s = d = e - E; }
  const float alpha = ebuf[e] / denom[d];
  for (int f = lane; f < C; f += 32)
    atomicAdd(acc + (size_t)d * C + f, alpha * H[(size_t)s * C + f]);
}

// ---- bias + ReLU (layers 0,1) --------------------------------------------
__global__ void gat_bias_relu(float* __restrict__ acc, const float* __restrict__ b,
                              int N, int C) {
  const int i = blockIdx.x * blockDim.x + threadIdx.x;
  if (i >= N * C) return;
  const float v = acc[i] + b[i % C];
  acc[i] = (v > 0.f) ? v : 0.f;
}

// ---- final: out = log_softmax(acc + b) over C=40 (one wave / node) -------
__global__ void gat_logsoftmax(const float* __restrict__ Z, const float* __restrict__ b,
                               float* __restrict__ out, int N, int C) {
  const int lane = threadIdx.x & 31;
  const int n = (blockIdx.x * blockDim.x + threadIdx.x) >> 5;
  if (n >= N) return;
  const int f2 = lane + 32;
  const float z1 = (lane < C) ? Z[(size_t)n * C + lane] + b[lane] : -FLT_MAX;
  const float z2 = (f2  < C) ? Z[(size_t)n * C + f2]  + b[f2]   : -FLT_MAX;
  float mx = fmaxf(z1, z2);
  for (int o = 16; o > 0; o >>= 1) mx = fmaxf(mx, __shfl_xor(mx, o, 32));
  float sum = 0.f;
  if (lane < C) sum += __expf(z1 - mx);
  if (f2  < C) sum += __expf(z2 - mx);
  for (int o = 16; o > 0; o >>= 1) sum += __shfl_xor(sum, o, 32);
  const float lse = mx + __logf(sum);
  if (lane < C) out[(size_t)n * C + lane] = z1 - lse;
  if (f2  < C) out[(size_t)n * C + f2]  = z2 - lse;
}

static inline int cdiv_i(long long a, long long b) { return (int)((a + b - 1) / b); }

extern "C" void kernel_launch(void* const* d_in, const int* in_sizes, int n_in,
                              void* d_out, int out_size, void* d_ws, size_t ws_size,
                              hipStream_t stream) {
  const float*     x  = (const float*)d_in[0];
  const long long* ei = (const long long*)d_in[1];   // reference dtype is int64
  const int D = 64;
  const int N = in_sizes[0] / D;                     // 100000
  const int E = in_sizes[1] / 2;                     // 1200000
  const int ET = E + N;                              // + virtual self-loops

  // workspace carve-up (~84 MB)
  float*    ws    = (float*)d_ws;
  float*    h     = ws;                              // N*64
  float*    acc0  = h    + (size_t)N * 64;           // N*64
  float*    acc1  = acc0 + (size_t)N * 64;           // N*64
  float*    as_   = acc1 + (size_t)N * 64;           // N
  float*    ad_   = as_  + N;                        // N
  float*    denom = ad_  + N;                        // N
  unsigned* mbits = (unsigned*)(denom + N);          // N
  float*    ebuf  = (float*)(mbits + N);             // E+N

  struct Layer { const float *W, *a_s, *a_d, *b; int C; };
  const Layer L[3] = {
    {(const float*)d_in[2],  (const float*)d_in[3],  (const float*)d_in[4],  (const float*)d_in[5],  64},
    {(const float*)d_in[6],  (const float*)d_in[7],  (const float*)d_in[8],  (const float*)d_in[9],  64},
    {(const float*)d_in[10], (const float*)d_in[11], (const float*)d_in[12], (const float*)d_in[13], 40},
  };
  const float* Xl[3] = { x, acc0, acc1 };
  float*       Al[3] = { acc0, acc1, acc0 };

  const int TB = 256;
  const int rtiles = cdiv_i(N, 16);
  const int gemm_blocks = cdiv_i((long long)rtiles * 32, TB);

  for (int l = 0; l < 3; ++l) {
    const int C = L[l].C;
    gat_init<<<cdiv_i((long long)N * C, TB), TB, 0, stream>>>(Al[l], denom, mbits, N, C);

    if (C == 64)
      gat_gemm_wmma<64, 64><<<gemm_blocks, TB, 0, stream>>>(Xl[l], L[l].W, h, N);
    else
      gat_gemm_wmma<64, 40><<<gemm_blocks, TB, 0, stream>>>(Xl[l], L[l].W, h, N);

    gat_alphas<<<cdiv_i((long long)N * 32, TB), TB, 0, stream>>>(h, L[l].a_s, L[l].a_d, as_, ad_, N, C);

    gat_edge_max<<<cdiv_i(ET, TB), TB, 0, stream>>>(ei, E, N, as_, ad_, ebuf, mbits);
    gat_edge_exp<<<cdiv_i(ET, TB), TB, 0, stream>>>(ei, E, N, mbits, ebuf, denom);
    gat_edge_agg<<<cdiv_i((long long)ET * 32, TB), TB, 0, stream>>>(ei, E, N, ebuf, denom, h, Al[l], C);

    if (l < 2)
      gat_bias_relu<<<cdiv_i((long long)N * C, TB), TB, 0, stream>>>(Al[l], L[l].b, N, C);
    else
      gat_logsoftmax<<<cdiv_i((long long)N * 32, TB), TB, 0, stream>>>(Al[l], L[l].b, (float*)d_out, N, C);
  }
}